// SimpleNet_36704790511896
// MI455X (gfx1250) — compile-verified
//
#include <hip/hip_runtime.h>

typedef __attribute__((ext_vector_type(2))) float v2f;
typedef __attribute__((ext_vector_type(8))) float v8f;

#define N_NODES 100000
#define N_EDGES 3200000
#define N_GRAPHS 64
#define NTILES 6250  // N_NODES / 16

__device__ __forceinline__ v8f wmma_f32(v2f a, v2f b, v8f c) {
  // D = A(16x4) * B(4x16) + C(16x16), all f32 (exact, no precision loss)
  return __builtin_amdgcn_wmma_f32_16x16x4_f32(false, a, false, b, (short)0, c,
                                               false, false);
}

// ---------------- utility kernels ----------------

__global__ void k_zero(float4* __restrict__ p, long n4) {
  long i = blockIdx.x * (long)blockDim.x + threadIdx.x;
  if (i < n4) p[i] = make_float4(0.f, 0.f, 0.f, 0.f);
}

__global__ void k_count(const int* __restrict__ dst, float* __restrict__ cnt) {
  int e = blockIdx.x * blockDim.x + threadIdx.x;
  if (e < N_EDGES) atomicAdd(&cnt[dst[e]], 1.0f);
}

__global__ void k_inv(const float* __restrict__ cnt, float* __restrict__ inv) {
  int n = blockIdx.x * blockDim.x + threadIdx.x;
  if (n < N_NODES) inv[n] = 1.0f / fmaxf(cnt[n], 1.0f);
}

// ---------------- WMMA GEMM kernels (f32 16x16x4) ----------------
// One wave per 16-row node tile. A: lane=(M)+(Khalf<<4), v2f={K,K+1}.
// B element (K,N): lane = N + 16*(K/2), vgpr = K%2. C/D: row v+8*(lane/16), col lane%16.

// out1[n][0:16] = x @ W1l ; out1[n][16:32] = x @ W1r   (K=64)
__global__ void k_gemm1(const float* __restrict__ x, const float* __restrict__ W1l,
                        const float* __restrict__ W1r, float* __restrict__ out1) {
  int lane = threadIdx.x & 31;
  int tile = blockIdx.x * (blockDim.x >> 5) + (threadIdx.x >> 5);
  if (tile >= NTILES) return;  // wave-uniform: EXEC all-1s inside
  int khalf = lane >> 4;
  int row = tile * 16 + (lane & 15);
  int n = lane & 15;
  const float* xr = x + (long)row * 64;
  v8f acc0 = {}; v8f acc1 = {};
#pragma unroll
  for (int k0 = 0; k0 < 64; k0 += 4) {
    int ka = k0 + 2 * khalf;
    v2f a;  a.x = xr[ka];              a.y = xr[ka + 1];
    v2f b0; b0.x = W1l[ka * 16 + n];   b0.y = W1l[(ka + 1) * 16 + n];
    v2f b1; b1.x = W1r[ka * 16 + n];   b1.y = W1r[(ka + 1) * 16 + n];
    acc0 = wmma_f32(a, b0, acc0);
    acc1 = wmma_f32(a, b1, acc1);
  }
  int m0 = tile * 16 + 8 * khalf;
#pragma unroll
  for (int v = 0; v < 8; v++) {
    out1[(long)(m0 + v) * 32 + n]      = acc0[v];
    out1[(long)(m0 + v) * 32 + 16 + n] = acc1[v];
  }
}

// h2 = (agg2*inv) @ W2l + b2 + h @ W2r   (two A-sources, K=16 each, N=32)
__global__ void k_gemm2(const float* __restrict__ agg2, const float* __restrict__ inv,
                        const float* __restrict__ h, const float* __restrict__ W2l,
                        const float* __restrict__ b2, const float* __restrict__ W2r,
                        float* __restrict__ h2) {
  int lane = threadIdx.x & 31;
  int tile = blockIdx.x * (blockDim.x >> 5) + (threadIdx.x >> 5);
  if (tile >= NTILES) return;
  int khalf = lane >> 4;
  int row = tile * 16 + (lane & 15);
  int n = lane & 15;
  float invn = inv[row];
  v8f acc0 = {}; v8f acc1 = {};
#pragma unroll
  for (int k0 = 0; k0 < 16; k0 += 4) {
    int ka = k0 + 2 * khalf;
    v2f a;  a.x = agg2[(long)row * 16 + ka] * invn;
            a.y = agg2[(long)row * 16 + ka + 1] * invn;
    v2f b0; b0.x = W2l[ka * 32 + n];        b0.y = W2l[(ka + 1) * 32 + n];
    v2f b1; b1.x = W2l[ka * 32 + 16 + n];   b1.y = W2l[(ka + 1) * 32 + 16 + n];
    acc0 = wmma_f32(a, b0, acc0);
    acc1 = wmma_f32(a, b1, acc1);
  }
#pragma unroll
  for (int k0 = 0; k0 < 16; k0 += 4) {
    int ka = k0 + 2 * khalf;
    v2f a;  a.x = h[(long)row * 16 + ka];   a.y = h[(long)row * 16 + ka + 1];
    v2f b0; b0.x = W2r[ka * 32 + n];        b0.y = W2r[(ka + 1) * 32 + n];
    v2f b1; b1.x = W2r[ka * 32 + 16 + n];   b1.y = W2r[(ka + 1) * 32 + 16 + n];
    acc0 = wmma_f32(a, b0, acc0);
    acc1 = wmma_f32(a, b1, acc1);
  }
  int m0 = tile * 16 + 8 * khalf;
#pragma unroll
  for (int v = 0; v < 8; v++) {
    h2[(long)(m0 + v) * 32 + n]      = acc0[v] + b2[n];
    h2[(long)(m0 + v) * 32 + 16 + n] = acc1[v] + b2[16 + n];
  }
}

// r = relu(h2); t3 = r @ W3l (21 cols), rr = r @ W3r (21 cols). K=32, N padded to 32.
__global__ void k_gemm3(const float* __restrict__ h2, const float* __restrict__ W3l,
                        const float* __restrict__ W3r, float* __restrict__ t3,
                        float* __restrict__ rr) {
  int lane = threadIdx.x & 31;
  int tile = blockIdx.x * (blockDim.x >> 5) + (threadIdx.x >> 5);
  if (tile >= NTILES) return;
  int khalf = lane >> 4;
  int row = tile * 16 + (lane & 15);
  int n = lane & 15;
  bool hi = (n < 5);  // column 16+n < 21
  v8f acc0 = {}, acc1 = {}, acc2 = {}, acc3 = {};
#pragma unroll
  for (int k0 = 0; k0 < 32; k0 += 4) {
    int ka = k0 + 2 * khalf;
    v2f a;  a.x = fmaxf(h2[(long)row * 32 + ka], 0.f);
            a.y = fmaxf(h2[(long)row * 32 + ka + 1], 0.f);
    v2f bl0; bl0.x = W3l[ka * 21 + n];  bl0.y = W3l[(ka + 1) * 21 + n];
    v2f bl1; bl1.x = hi ? W3l[ka * 21 + 16 + n] : 0.f;
             bl1.y = hi ? W3l[(ka + 1) * 21 + 16 + n] : 0.f;
    v2f br0; br0.x = W3r[ka * 21 + n];  br0.y = W3r[(ka + 1) * 21 + n];
    v2f br1; br1.x = hi ? W3r[ka * 21 + 16 + n] : 0.f;
             br1.y = hi ? W3r[(ka + 1) * 21 + 16 + n] : 0.f;
    acc0 = wmma_f32(a, bl0, acc0);
    acc1 = wmma_f32(a, bl1, acc1);
    acc2 = wmma_f32(a, br0, acc2);
    acc3 = wmma_f32(a, br1, acc3);
  }
  int m0 = tile * 16 + 8 * khalf;
#pragma unroll
  for (int v = 0; v < 8; v++) {
    long base = (long)(m0 + v) * 21;
    t3[base + n] = acc0[v];
    rr[base + n] = acc2[v];
    if (hi) { t3[base + 16 + n] = acc1[v]; rr[base + 16 + n] = acc3[v]; }
  }
}

// ---------------- edge scatter (L2-resident float atomics) ----------------

__global__ void k_scatter16(const int* __restrict__ src, const int* __restrict__ dst,
                            const float* __restrict__ t, int tstride,
                            float* __restrict__ agg) {
  long i = blockIdx.x * (long)blockDim.x + threadIdx.x;
  long e = i >> 4; int f = (int)(i & 15);
  if (e < N_EDGES) {
    int s = src[e], d = dst[e];
    atomicAdd(&agg[(long)d * 16 + f], t[(long)s * tstride + f]);
  }
}

__global__ void k_scatter21(const int* __restrict__ src, const int* __restrict__ dst,
                            const float* __restrict__ t3, float* __restrict__ agg3) {
  long i = blockIdx.x * (long)blockDim.x + threadIdx.x;
  long e = i >> 5; int f = (int)(i & 31);
  if (e < N_EDGES && f < 21) {
    int s = src[e], d = dst[e];
    atomicAdd(&agg3[(long)d * 21 + f], t3[(long)s * 21 + f]);
  }
}

// ---------------- elementwise / pooling / classifier ----------------

__global__ void k_e1(const float* __restrict__ agg1, const float* __restrict__ inv,
                     const float* __restrict__ b1, const float* __restrict__ out1,
                     float* __restrict__ h) {
  int i = blockIdx.x * blockDim.x + threadIdx.x;
  if (i < N_NODES * 16) {
    int n = i >> 4, j = i & 15;
    h[i] = fmaxf(agg1[i] * inv[n] + b1[j] + out1[(long)n * 32 + 16 + j], 0.0f);
  }
}

__global__ void k_pool(const float* __restrict__ h2, const int* __restrict__ batch,
                       float* __restrict__ gsum, float* __restrict__ gcnt) {
  long i = blockIdx.x * (long)blockDim.x + threadIdx.x;
  int n = (int)(i >> 5), f = (int)(i & 31);
  if (n < N_NODES) {
    int b = batch[n];
    atomicAdd(&gsum[b * 32 + f], h2[(long)n * 32 + f]);
    if (f == 0) atomicAdd(&gcnt[b], 1.0f);
  }
}

__global__ void k_e3(const float* __restrict__ agg3, const float* __restrict__ inv,
                     const float* __restrict__ b3, const float* __restrict__ rr,
                     float* __restrict__ color) {
  long i = blockIdx.x * (long)blockDim.x + threadIdx.x;
  if (i < (long)N_NODES * 21) {
    int n = (int)(i / 21), c = (int)(i % 21);
    color[i] = agg3[i] * inv[n] + b3[c] + rr[i];
  }
}

__global__ void k_classif(const float* __restrict__ gsum, const float* __restrict__ gcnt,
                          const float* __restrict__ Wc, const float* __restrict__ bc,
                          float* __restrict__ out) {
  int i = blockIdx.x * blockDim.x + threadIdx.x;
  if (i < N_GRAPHS * 10) {
    int g = i / 10, c = i % 10;
    float s = 1.0f / fmaxf(gcnt[g], 1.0f);
    float acc = bc[c];
#pragma unroll
    for (int k = 0; k < 32; k++) acc += gsum[g * 32 + k] * s * Wc[k * 10 + c];
    out[i] = acc;
  }
}

// ---------------- launch ----------------

extern "C" void kernel_launch(void* const* d_in, const int* in_sizes, int n_in,
                              void* d_out, int out_size, void* d_ws, size_t ws_size,
                              hipStream_t stream) {
  (void)in_sizes; (void)n_in; (void)out_size; (void)ws_size;
  const float* x   = (const float*)d_in[0];
  const int*   ei  = (const int*)d_in[1];
  const int*   src = ei;
  const int*   dst = ei + N_EDGES;
  const int* batch = (const int*)d_in[2];
  const float* W1l = (const float*)d_in[3];
  const float* b1  = (const float*)d_in[4];
  const float* W1r = (const float*)d_in[5];
  const float* W2l = (const float*)d_in[6];
  const float* b2  = (const float*)d_in[7];
  const float* W2r = (const float*)d_in[8];
  const float* W3l = (const float*)d_in[9];
  const float* b3  = (const float*)d_in[10];
  const float* W3r = (const float*)d_in[11];
  const float* Wc  = (const float*)d_in[12];
  const float* bc  = (const float*)d_in[13];
  float* classif = (float*)d_out;          // 64 x 10
  float* color   = (float*)d_out + 640;    // 100000 x 21

  // workspace partition (floats); accumulators first (contiguous zero region)
  float* ws   = (float*)d_ws;
  float* agg1 = ws;                  // 1,600,000
  float* agg2 = agg1 + 1600000;      // 1,600,000
  float* agg3 = agg2 + 1600000;      // 2,100,000
  float* cnt  = agg3 + 2100000;      //   100,000
  float* gsum = cnt + 100000;        //     2,048
  float* gcnt = gsum + 2048;         //        64
  const long ZERON = 1600000L + 1600000L + 2100000L + 100000L + 2048L + 64L; // 5,402,112
  float* out1 = gcnt + 64;           // 3,200,000
  float* h    = out1 + 3200000;      // 1,600,000
  float* h2   = h + 1600000;         // 3,200,000
  float* t3   = h2 + 3200000;        // 2,100,000
  float* rr   = t3 + 2100000;        // 2,100,000
  float* inv  = rr + 2100000;        //   100,000

  const int B = 256;
  // 1. zero accumulators (ZERON divisible by 4)
  k_zero<<<(int)((ZERON / 4 + B - 1) / B), B, 0, stream>>>((float4*)ws, ZERON / 4);
  // 2-3. degree counts and reciprocals
  k_count<<<(N_EDGES + B - 1) / B, B, 0, stream>>>(dst, cnt);
  k_inv<<<(N_NODES + B - 1) / B, B, 0, stream>>>(cnt, inv);
  // 4. layer-1 node transforms (WMMA)
  const int GB = (NTILES + 7) / 8;  // 8 waves per 256-thread block
  k_gemm1<<<GB, B, 0, stream>>>(x, W1l, W1r, out1);
  // 5. scatter y1 (16 floats/edge), 6. combine+relu -> h
  k_scatter16<<<(int)(((long)N_EDGES * 16 + B - 1) / B), B, 0, stream>>>(src, dst, out1, 32, agg1);
  k_e1<<<(N_NODES * 16 + B - 1) / B, B, 0, stream>>>(agg1, inv, b1, out1, h);
  // 7. scatter h, 8. layer-2 GEMM (WMMA)
  k_scatter16<<<(int)(((long)N_EDGES * 16 + B - 1) / B), B, 0, stream>>>(src, dst, h, 16, agg2);
  k_gemm2<<<GB, B, 0, stream>>>(agg2, inv, h, W2l, b2, W2r, h2);
  // 9. graph mean-pool atomics (pre-relu h2)
  k_pool<<<(int)(((long)N_NODES * 32 + B - 1) / B), B, 0, stream>>>(h2, batch, gsum, gcnt);
  // 10. layer-3 transforms (WMMA, relu on the fly), 11. scatter 21 floats/edge
  k_gemm3<<<GB, B, 0, stream>>>(h2, W3l, W3r, t3, rr);
  k_scatter21<<<(int)(((long)N_EDGES * 32 + B - 1) / B), B, 0, stream>>>(src, dst, t3, agg3);
  // 12. color output, 13. classifier output
  k_e3<<<(int)(((long)N_NODES * 21 + B - 1) / B), B, 0, stream>>>(agg3, inv, b3, rr, color);
  k_classif<<<(N_GRAPHS * 10 + B - 1) / B, B, 0, stream>>>(gsum, gcnt, Wc, bc, classif);
}